// FlowAlignedSmoothingEffect_47012712022499
// MI455X (gfx1250) — compile-verified
//
#include <hip/hip_runtime.h>

// ---------------------------------------------------------------------------
// Flow-aligned smoothing (LIC-style) for B=2, C=3, H=W=1024 fp32.
// Strategy: 32x32 output tile per 256-thread block; stage a 44x44 region
// (halo=6 covers max 4px march + 1px bilinear) of 3 image + 2 tangent
// channels into LDS via gfx1250 async global->LDS copies, then do the whole
// 8-step march with ds_load_b128 / ds_load_b64 bilinear taps from LDS.
// ---------------------------------------------------------------------------

#define TILE     32
#define HALO     6
#define RW       (TILE + 2 * HALO)   // 44
#define RNUM     (RW * RW)           // 1936 texels
#define NTHREADS 256

constexpr int   BATCH   = 2;
constexpr int   CH      = 3;
constexpr int   H       = 1024;
constexpr int   W       = 1024;
constexpr int   HW      = H * W;
constexpr int   N_STEPS = 4;
constexpr float STEP_SZ = 0.3333f;

// ---- gfx1250 async global->LDS copy (ASYNCcnt path), with safe fallback ----
#if defined(__HIP_DEVICE_COMPILE__) && __has_builtin(__builtin_amdgcn_global_load_async_to_lds_b32)
#define HAVE_ASYNC_LDS 1
#define ASYNC_COPY_B32(gsrc, ldst)                                                   \
  __builtin_amdgcn_global_load_async_to_lds_b32(                                     \
      (__attribute__((address_space(1))) int*)(unsigned long long)(gsrc),            \
      (__attribute__((address_space(3))) int*)(unsigned long long)(ldst), 0, 0)
#else
#define ASYNC_COPY_B32(gsrc, ldst) (*(ldst) = *(gsrc))
#endif

__device__ __forceinline__ void async_fill_wait() {
#if defined(HAVE_ASYNC_LDS)
#if __has_builtin(__builtin_amdgcn_s_wait_asynccnt)
  __builtin_amdgcn_s_wait_asynccnt(0);
#else
  asm volatile("s_wait_asynccnt 0" ::: "memory");
#endif
#endif
}

__device__ __forceinline__ int clampi(int v, int lo, int hi) {
  return v < lo ? lo : (v > hi ? hi : v);
}

__global__ __launch_bounds__(NTHREADS)
void flow_aligned_smoothing_kernel(const float* __restrict__ x,
                                   const float* __restrict__ tangent,
                                   const float* __restrict__ sigma,
                                   float* __restrict__ out) {
  // Interleaved LDS tiles: image as float4 (xyz used, w pad -> b128 taps),
  // tangent as float2 (-> b64 taps). Total 46,464 B.
  __shared__ float4 s_img[RNUM];
  __shared__ float2 s_tan[RNUM];

  const int t       = threadIdx.x;
  const int tile_x0 = blockIdx.x * TILE;
  const int tile_y0 = blockIdx.y * TILE;
  const int b       = blockIdx.z;
  const int rx0     = tile_x0 - HALO;
  const int ry0     = tile_y0 - HALO;

  const float* xb = x + (size_t)b * CH * HW;
  const float* tb = tangent + (size_t)b * 2 * HW;

  // ---- Stage region into LDS (clamp-to-edge per lane; async, no VGPR trip).
  for (int idx = t; idx < RNUM; idx += NTHREADS) {
    const int ry = idx / RW;
    const int rx = idx - ry * RW;
    const int gy = clampi(ry0 + ry, 0, H - 1);
    const int gx = clampi(rx0 + rx, 0, W - 1);
    const size_t g = (size_t)gy * W + gx;
    float* id = (float*)&s_img[idx];
    float* td = (float*)&s_tan[idx];
    ASYNC_COPY_B32(xb + g,            id + 0);
    ASYNC_COPY_B32(xb + HW + g,       id + 1);
    ASYNC_COPY_B32(xb + 2 * HW + g,   id + 2);
    ASYNC_COPY_B32(tb + g,            td + 0);
    ASYNC_COPY_B32(tb + HW + g,       td + 1);
  }
  async_fill_wait();
  __syncthreads();

  // ---- Per-batch Gaussian weights (uniform across pixels; hoisted).
  const float size_factor = (float)(H < W ? H : W) / 1024.0f;
  const float sig         = sigma[b] * size_factor;
  const float half_width  = 2.0f * sig;
  const float inv_2s2     = 1.0f / (2.0f * sig * sig);
  const float step        = 1.0f / (STEP_SZ * size_factor);
  float kbase[N_STEPS];
#pragma unroll
  for (int it = 0; it < N_STEPS; ++it) {
    const float r = (float)(it + 1) * step;
    kbase[it] = (r < half_width) ? __expf(-r * r * inv_2s2) : 0.0f;
  }

  constexpr float invW = 1.0f / (float)W;
  constexpr float invH = 1.0f / (float)H;

  const int tx  = t & (TILE - 1);
  const int ty0 = t >> 5;  // 0..7; each thread owns 4 rows (stride 8)

#pragma unroll
  for (int kk = 0; kk < 4; ++kk) {
    const int iy = ty0 + kk * 8;
    const int j  = tile_x0 + tx;   // global x
    const int i  = tile_y0 + iy;   // global y

    const int    lc = (iy + HALO) * RW + (tx + HALO);
    const float4 xc = s_img[lc];
    const float2 tc = s_tan[lc];

    const float psx = ((float)j + 0.5f) * invW;
    const float psy = ((float)i + 0.5f) * invH;

    float accx = 0.f, accy = 0.f, accz = 0.f, wsum = 0.f;

#pragma unroll
    for (int s = 0; s < 2; ++s) {               // +tangent then -tangent march
      float vx = s ? -tc.x : tc.x;
      float vy = s ? -tc.y : tc.y;
      float px = psx + vx * invW;
      float py = psy + vy * invH;

#pragma unroll
      for (int it = 0; it < N_STEPS; ++it) {
        float k = kbase[it];
        if (!(px >= 0.f && px < 1.f && py >= 0.f && py < 1.f)) k = 0.f;

        // Bilinear taps (align_corners=False convention, border clamp).
        const float fx  = px * (float)W - 0.5f;
        const float fy  = py * (float)H - 0.5f;
        const float x0f = floorf(fx);
        const float y0f = floorf(fy);
        const float wx  = fx - x0f;
        const float wy  = fy - y0f;
        const int x0 = clampi((int)x0f, 0, W - 1);
        const int x1 = (x0 + 1 < W) ? x0 + 1 : W - 1;
        const int y0 = clampi((int)y0f, 0, H - 1);
        const int y1 = (y0 + 1 < H) ? y0 + 1 : H - 1;

        const int r0  = (y0 - ry0) * RW;
        const int r1  = (y1 - ry0) * RW;
        const int c0  = x0 - rx0;
        const int c1  = x1 - rx0;
        const int l00 = r0 + c0, l01 = r0 + c1, l10 = r1 + c0, l11 = r1 + c1;

        const float w00 = (1.f - wx) * (1.f - wy);
        const float w01 = wx * (1.f - wy);
        const float w10 = (1.f - wx) * wy;
        const float w11 = wx * wy;

        const float4 a00 = s_img[l00], a01 = s_img[l01];
        const float4 a10 = s_img[l10], a11 = s_img[l11];
        const float2 b00 = s_tan[l00], b01 = s_tan[l01];
        const float2 b10 = s_tan[l10], b11 = s_tan[l11];

        accx += k * (a00.x * w00 + a01.x * w01 + a10.x * w10 + a11.x * w11);
        accy += k * (a00.y * w00 + a01.y * w01 + a10.y * w10 + a11.y * w11);
        accz += k * (a00.z * w00 + a01.z * w01 + a10.z * w10 + a11.z * w11);
        wsum += k;

        float tfx = b00.x * w00 + b01.x * w01 + b10.x * w10 + b11.x * w11;
        float tfy = b00.y * w00 + b01.y * w01 + b10.y * w10 + b11.y * w11;
        const float vt = vx * tfx + vy * tfy;
        if (vt < 0.f) { tfx = -tfx; tfy = -tfy; }
        px += tfx * invW;
        py += tfy * invH;
        vx = tfx;
        vy = tfy;
      }
    }

    const float inv = 1.0f / (1.0f + wsum);
    const size_t o  = (size_t)b * CH * HW + (size_t)i * W + j;
    out[o]          = (xc.x + accx) * inv;
    out[o + HW]     = (xc.y + accy) * inv;
    out[o + 2 * HW] = (xc.z + accz) * inv;
  }
}

extern "C" void kernel_launch(void* const* d_in, const int* in_sizes, int n_in,
                              void* d_out, int out_size, void* d_ws, size_t ws_size,
                              hipStream_t stream) {
  (void)in_sizes; (void)n_in; (void)out_size; (void)d_ws; (void)ws_size;
  const float* x       = (const float*)d_in[0];
  const float* tangent = (const float*)d_in[1];
  const float* sigma   = (const float*)d_in[2];
  float* out           = (float*)d_out;

  dim3 grid(W / TILE, H / TILE, BATCH);
  dim3 block(NTHREADS);
  flow_aligned_smoothing_kernel<<<grid, block, 0, stream>>>(x, tangent, sigma, out);
}